// PC_shielded_electrostatics_24172075942131
// MI455X (gfx1250) — compile-verified
//
#include <hip/hip_runtime.h>

// PhysNet-style shielded point-charge electrostatics with Poly6 switch +
// segment-sum over idx_i. Memory-bound scatter/gather: stream 6.4M
// (dist, idx_i, idx_j) triples with non-temporal b128 loads, gather charges
// from an L2-resident 400KB table, scatter energies with native
// (non-returning) global_atomic_add_f32 at device scope.

#define PC_CUTOFF     10.0f
#define PC_CUTOFF_SR  2.0f
#define PC_KEHALF     7.199822675975274f

typedef float f32x4 __attribute__((ext_vector_type(4)));
typedef int   i32x4 __attribute__((ext_vector_type(4)));

// Non-returning fp32 atomic add at device scope (L2 atomic units).
// Inline asm guarantees the hardware instruction (no CAS-loop expansion).
__device__ __forceinline__ void atomic_add_f32_dev(float* addr, float v) {
    asm volatile("global_atomic_add_f32 %0, %1, off scope:SCOPE_DEV"
                 :
                 : "v"(addr), "v"(v)
                 : "memory");
}

__device__ __forceinline__ float pair_energy(float d, float qi, float qj) {
    // 1/d via v_rcp_f32 (~1 ulp), 1/sqrt(d^2+1) via v_rsq_f32.
    const float inv_d = __builtin_amdgcn_rcpf(d);
    const float t     = __builtin_fmaf(d, d, 1.0f);   // d^2 + 1
    const float rs    = __builtin_amdgcn_rsqf(t);     // 1/sqrt(t)
    const float s     = t * rs;                       // sqrt(d^2+1) (shielded)

    const float inv_cut2     = 1.0f / (PC_CUTOFF * PC_CUTOFF);  // 0.01
    const float two_over_cut = 2.0f / PC_CUTOFF;                // 0.2

    const float E_ord = __builtin_fmaf(d, inv_cut2, inv_d) - two_over_cut;
    const float E_shd = __builtin_fmaf(s, inv_cut2, rs)    - two_over_cut;

    // Poly6 switch: p = 1 - 10x^3 + 15x^4 - 6x^5,  x = d / rc_sr
    const float x  = d * (1.0f / PC_CUTOFF_SR);
    const float x3 = x * x * x;
    const float p  = __builtin_fmaf(
        x3, __builtin_fmaf(x, __builtin_fmaf(-6.0f, x, 15.0f), -10.0f), 1.0f);
    const float sw_off = (d < PC_CUTOFF_SR) ? p : 0.0f;

    const float blend = __builtin_fmaf(sw_off, E_shd - E_ord, E_ord);
    const float E = PC_KEHALF * qi * qj * blend;
    return (d <= PC_CUTOFF) ? E : 0.0f;
}

__global__ __launch_bounds__(256) void pc_pair_kernel(
    const float* __restrict__ q,
    const float* __restrict__ dist,
    const int*   __restrict__ idx_i,
    const int*   __restrict__ idx_j,
    float*       __restrict__ out,
    int n_pairs)
{
    const int tid  = blockIdx.x * blockDim.x + threadIdx.x;
    const int base = tid * 4;
    if (base >= n_pairs) return;

    if (base + 3 < n_pairs) {
        // Non-temporal b128 streaming loads: 4 pairs per thread, read-once
        // data stays out of the hot near-caches (q table + out are the
        // reused working set).
        const f32x4 d4 = __builtin_nontemporal_load(
            reinterpret_cast<const f32x4*>(dist + base));
        const i32x4 i4 = __builtin_nontemporal_load(
            reinterpret_cast<const i32x4*>(idx_i + base));
        const i32x4 j4 = __builtin_nontemporal_load(
            reinterpret_cast<const i32x4*>(idx_j + base));

        const float e0 = pair_energy(d4.x, q[i4.x], q[j4.x]);
        const float e1 = pair_energy(d4.y, q[i4.y], q[j4.y]);
        const float e2 = pair_energy(d4.z, q[i4.z], q[j4.z]);
        const float e3 = pair_energy(d4.w, q[i4.w], q[j4.w]);

        atomic_add_f32_dev(out + i4.x, e0);
        atomic_add_f32_dev(out + i4.y, e1);
        atomic_add_f32_dev(out + i4.z, e2);
        atomic_add_f32_dev(out + i4.w, e3);
    } else {
        // Tail (only reachable if n_pairs % 4 != 0).
        for (int k = base; k < n_pairs; ++k) {
            const float e = pair_energy(dist[k], q[idx_i[k]], q[idx_j[k]]);
            atomic_add_f32_dev(out + idx_i[k], e);
        }
    }
}

__global__ __launch_bounds__(256) void pc_zero_kernel(float* __restrict__ out, int n) {
    const int i = blockIdx.x * blockDim.x + threadIdx.x;
    if (i < n) out[i] = 0.0f;
}

extern "C" void kernel_launch(void* const* d_in, const int* in_sizes, int n_in,
                              void* d_out, int out_size, void* d_ws, size_t ws_size,
                              hipStream_t stream) {
    (void)n_in; (void)d_ws; (void)ws_size; (void)out_size;

    const float* q    = (const float*)d_in[0];  // atomic_charges  [N_ATOMS]
    const float* dist = (const float*)d_in[1];  // distances       [N_PAIRS]
    const int*   ii   = (const int*)  d_in[2];  // idx_i           [N_PAIRS]
    const int*   jj   = (const int*)  d_in[3];  // idx_j           [N_PAIRS]
    float*       out  = (float*)d_out;          // per-atom energy [N_ATOMS]

    const int n_atoms = in_sizes[0];
    const int n_pairs = in_sizes[1];

    // 1) Zero the output (harness poisons d_out; atomics need a clean base).
    {
        const int threads = 256;
        const int blocks  = (n_atoms + threads - 1) / threads;
        pc_zero_kernel<<<blocks, threads, 0, stream>>>(out, n_atoms);
    }

    // 2) Pair energies + segment-sum via device-scope fp32 atomics.
    //    4 pairs/thread => b128 loads; 6.4M pairs -> exactly 6250 blocks.
    {
        const int threads  = 256;
        const int nthreads = (n_pairs + 3) / 4;
        const int blocks   = (nthreads + threads - 1) / threads;
        pc_pair_kernel<<<blocks, threads, 0, stream>>>(q, dist, ii, jj, out, n_pairs);
    }
}